// RuleAwareGraphConv_10806137716887
// MI455X (gfx1250) — compile-verified
//
#include <hip/hip_runtime.h>
#include <hip/hip_bf16.h>

typedef __attribute__((ext_vector_type(2))) float v2f;
typedef __attribute__((ext_vector_type(8))) float v8f;

#define IN_DIM 64
#define OUT_DIM 64
#define N_REL 8
#define N_ACTIVE 4
#define SCALE 0.125f /* 1/sqrt(64) */

// ---------- helpers: order-preserving float<->uint encoding for atomicMax ----------
__device__ __forceinline__ unsigned encf(float f) {
  unsigned u = __float_as_uint(f);
  return (u & 0x80000000u) ? ~u : (u | 0x80000000u);
}
__device__ __forceinline__ float decf(unsigned u) {
  unsigned v = (u & 0x80000000u) ? (u & 0x7FFFFFFFu) : ~u;
  return __uint_as_float(v);
}

// ---------- zero init ----------
__global__ void zero_kernel(float* __restrict__ p, int n) {
  int i = blockIdx.x * blockDim.x + threadIdx.x;
  if (i < n) p[i] = 0.0f;
}

// ---------- tiny precompute: rel means, relk (incl Wk_bias), key_rule ----------
__global__ void precompute_kernel(const float* __restrict__ W_r,
                                  const float* __restrict__ Wk_w,
                                  const float* __restrict__ Wk_b,
                                  const float* __restrict__ rule_emb,
                                  const int* __restrict__ rule_ids,
                                  float* __restrict__ relk,       // [8*64]
                                  float* __restrict__ key_rule) { // [4*64]
  __shared__ float s_rel[N_REL * IN_DIM];
  int tid = threadIdx.x;
  if (tid < 512) {  // rel_emb_table[t][k] = mean_n W_r[t][k][n]
    int t = tid >> 6, k = tid & 63;
    const float* p = W_r + ((size_t)t * IN_DIM + k) * OUT_DIM;
    float s = 0.f;
    for (int n = 0; n < OUT_DIM; ++n) s += p[n];
    s_rel[tid] = s * (1.0f / OUT_DIM);
  }
  __syncthreads();
  if (tid < 512) {  // relk[t][n] = rel_emb_table[t] @ Wk2 + Wk_bias
    int t = tid >> 6, n = tid & 63;
    float s = Wk_b[n];
    for (int k = 0; k < IN_DIM; ++k)
      s += s_rel[t * IN_DIM + k] * Wk_w[(size_t)(IN_DIM + k) * OUT_DIM + n];
    relk[tid] = s;
  }
  if (tid < 256) {  // key_rule[j][n] = rule_emb[rule_ids[j]] @ Wk3
    int j = tid >> 6, n = tid & 63;
    int rid = rule_ids[j];
    float s = 0.f;
    for (int k = 0; k < IN_DIM; ++k)
      s += rule_emb[(size_t)rid * IN_DIM + k] * Wk_w[(size_t)(2 * IN_DIM + k) * OUT_DIM + n];
    key_rule[tid] = s;
  }
}

// ---------- fp32 WMMA node GEMM: 10 weight matrices (q, xk, xw[0..7]) ----------
// One wave -> 16 rows x 64 cols.  A: 16x4 f32 frag (v2f/lane), B: 4x16 (v2f/lane),
// C/D: 16x16 (v8f/lane).  16 K-steps x 4 N-tiles = 64 v_wmma_f32_16x16x4_f32 / tile.
__global__ __launch_bounds__(256) void gemm64_wmma_kernel(
    const float* __restrict__ x,
    const float* __restrict__ Wq_w, const float* __restrict__ Wq_b,
    const float* __restrict__ Wk_w, const float* __restrict__ W_r,
    float* __restrict__ ws_q, float* __restrict__ ws_xk, float* __restrict__ ws_xw,
    int n_rows) {
  int sel = blockIdx.y;  // 0: q (+bias), 1: xk, 2..9: xw[sel-2]
  const float* W;
  const float* bias = nullptr;
  float* out;
  if (sel == 0)      { W = Wq_w; bias = Wq_b; out = ws_q; }
  else if (sel == 1) { W = Wk_w;              out = ws_xk; }  // first 64 rows = Wk1
  else { W = W_r + (size_t)(sel - 2) * IN_DIM * OUT_DIM;
         out = ws_xw + (size_t)(sel - 2) * (size_t)n_rows * OUT_DIM; }

  __shared__ float sW[IN_DIM * OUT_DIM];  // 16 KB
  for (int i = threadIdx.x; i < IN_DIM * OUT_DIM; i += 256) sW[i] = W[i];
  __syncthreads();

  int wave = threadIdx.x >> 5, lane = threadIdx.x & 31;
  int half = lane & 15, hi = lane >> 4;
  int tileRow = (blockIdx.x * 8 + wave) * 16;
  if (tileRow >= n_rows) return;  // wave-uniform exit (EXEC stays all-ones)

  v8f acc[4];
#pragma unroll
  for (int nt = 0; nt < 4; ++nt) {
    float bv = bias ? bias[nt * 16 + half] : 0.0f;  // C: N depends only on lane
    acc[nt] = (v8f){bv, bv, bv, bv, bv, bv, bv, bv};
  }

  int rowA = tileRow + half;
  if (rowA >= n_rows) rowA = n_rows - 1;  // clamp loads, keep EXEC full
  const float* xr = x + (size_t)rowA * IN_DIM;

  for (int kk = 0; kk < 16; ++kk) {
    int kb = kk * 4 + hi * 2;  // lanes 0-15 hold K=4kk+{0,1}; lanes 16-31 K=4kk+{2,3}
    v2f a;
    a.x = xr[kb];
    a.y = xr[kb + 1];
#pragma unroll
    for (int nt = 0; nt < 4; ++nt) {
      v2f b;
      b.x = sW[kb * OUT_DIM + nt * 16 + half];
      b.y = sW[(kb + 1) * OUT_DIM + nt * 16 + half];
      acc[nt] = __builtin_amdgcn_wmma_f32_16x16x4_f32(
          false, a, false, b, (short)0, acc[nt], false, false);
    }
  }

#pragma unroll
  for (int nt = 0; nt < 4; ++nt) {
#pragma unroll
    for (int v = 0; v < 8; ++v) {  // D: lanes 0-15 -> M=v, lanes 16-31 -> M=v+8
      int row = tileRow + v + 8 * hi;
      if (row < n_rows) out[(size_t)row * OUT_DIM + nt * 16 + half] = acc[nt][v];
    }
  }
}

// ---------- per-node rule scores: qr[n][j] = q[n] . key_rule[j] ----------
__global__ void qr_kernel(const float* __restrict__ q, const float* __restrict__ key_rule,
                          float* __restrict__ qr, int n_nodes) {
  __shared__ float kr[N_ACTIVE * OUT_DIM];
  if (threadIdx.x < N_ACTIVE * OUT_DIM) kr[threadIdx.x] = key_rule[threadIdx.x];
  __syncthreads();
  int n = blockIdx.x * blockDim.x + threadIdx.x;
  if (n >= n_nodes) return;
  float a0 = 0, a1 = 0, a2 = 0, a3 = 0;
  const float* qp = q + (size_t)n * OUT_DIM;
#pragma unroll 16
  for (int k = 0; k < OUT_DIM; ++k) {
    float qv = qp[k];
    a0 += qv * kr[0 * OUT_DIM + k];
    a1 += qv * kr[1 * OUT_DIM + k];
    a2 += qv * kr[2 * OUT_DIM + k];
    a3 += qv * kr[3 * OUT_DIM + k];
  }
  float* o = qr + (size_t)n * 4;
  o[0] = a0; o[1] = a1; o[2] = a2; o[3] = a3;
}

// ---------- edge pass 1: scores + segment max ----------
__global__ void score_kernel(const int* __restrict__ ei, const int* __restrict__ et,
                             const float* __restrict__ q, const float* __restrict__ xk,
                             const float* __restrict__ relk, const float* __restrict__ qr,
                             float* __restrict__ escore, unsigned* __restrict__ segmax,
                             int n_edges) {
  __shared__ float s_relk[N_REL * OUT_DIM];
  for (int i = threadIdx.x; i < N_REL * OUT_DIM; i += blockDim.x) s_relk[i] = relk[i];
  __syncthreads();
  int e = blockIdx.x * blockDim.x + threadIdx.x;
  if (e >= n_edges) return;
  int src = ei[e], dst = ei[n_edges + e], t = et[e];
  const float* qp = q + (size_t)dst * OUT_DIM;
  const float* kp = xk + (size_t)src * OUT_DIM;
  const float* rp = s_relk + t * OUT_DIM;
  float sb = 0.f;
#pragma unroll 16
  for (int k = 0; k < OUT_DIM; ++k) sb += qp[k] * (kp[k] + rp[k]);
  const float* qrr = qr + (size_t)dst * 4;
#pragma unroll
  for (int j = 0; j < N_ACTIVE; ++j) {
    float sc = (sb + qrr[j]) * SCALE;
    escore[(size_t)e * 4 + j] = sc;
    atomicMax(&segmax[(size_t)dst * 4 + j], encf(sc));
  }
}

// ---------- edge pass 2: exp + segment sum ----------
__global__ void expsum_kernel(const int* __restrict__ ei, const unsigned* __restrict__ segmax,
                              float* __restrict__ escore, float* __restrict__ segsum,
                              int n_edges) {
  int e = blockIdx.x * blockDim.x + threadIdx.x;
  if (e >= n_edges) return;
  int dst = ei[n_edges + e];
#pragma unroll
  for (int j = 0; j < N_ACTIVE; ++j) {
    float m = decf(segmax[(size_t)dst * 4 + j]);
    float ev = __expf(escore[(size_t)e * 4 + j] - m);
    escore[(size_t)e * 4 + j] = ev;
    atomicAdd(&segsum[(size_t)dst * 4 + j], ev);
  }
}

// ---------- edge pass 3: mean attention * gathered xw row -> scatter-add ----------
__global__ void accum_kernel(const int* __restrict__ ei, const int* __restrict__ et,
                             const float* __restrict__ escore, const float* __restrict__ segsum,
                             const float* __restrict__ xw, float* __restrict__ out_acc,
                             int n_edges, int n_nodes) {
  int e = blockIdx.x * blockDim.x + threadIdx.x;
  if (e >= n_edges) return;
  int src = ei[e], dst = ei[n_edges + e], t = et[e];
  float ma = 0.f;
#pragma unroll
  for (int j = 0; j < N_ACTIVE; ++j)
    ma += escore[(size_t)e * 4 + j] / segsum[(size_t)dst * 4 + j];
  ma *= 0.25f;
  const float* m = xw + ((size_t)t * n_nodes + src) * OUT_DIM;
  float* o = out_acc + (size_t)dst * OUT_DIM;
#pragma unroll 8
  for (int k = 0; k < OUT_DIM; ++k) atomicAdd(&o[k], ma * m[k]);
}

// ---------- finalize: +bias, LayerNorm, ReLU (one wave per node) ----------
__global__ void finalize_kernel(const float* __restrict__ acc, const float* __restrict__ bias,
                                const float* __restrict__ g, const float* __restrict__ b,
                                float* __restrict__ out, int n_nodes) {
  int wid = (blockIdx.x * blockDim.x + threadIdx.x) >> 5;
  int lane = threadIdx.x & 31;
  if (wid >= n_nodes) return;
  float v0 = acc[(size_t)wid * 64 + lane] + bias[lane];
  float v1 = acc[(size_t)wid * 64 + 32 + lane] + bias[32 + lane];
  float s = v0 + v1;
#pragma unroll
  for (int off = 16; off; off >>= 1) s += __shfl_xor(s, off, 32);
  float mu = s * (1.0f / 64.0f);
  float d0 = v0 - mu, d1 = v1 - mu;
  float vs = d0 * d0 + d1 * d1;
#pragma unroll
  for (int off = 16; off; off >>= 1) vs += __shfl_xor(vs, off, 32);
  float inv = rsqrtf(vs * (1.0f / 64.0f) + 1e-5f);
  out[(size_t)wid * 64 + lane]      = fmaxf(d0 * inv * g[lane] + b[lane], 0.0f);
  out[(size_t)wid * 64 + 32 + lane] = fmaxf(d1 * inv * g[32 + lane] + b[32 + lane], 0.0f);
}

extern "C" void kernel_launch(void* const* d_in, const int* in_sizes, int n_in,
                              void* d_out, int out_size, void* d_ws, size_t ws_size,
                              hipStream_t stream) {
  const float* x        = (const float*)d_in[0];
  const int*   ei       = (const int*)d_in[1];
  const int*   et       = (const int*)d_in[2];
  const int*   rule_ids = (const int*)d_in[3];
  const float* W_r      = (const float*)d_in[4];
  const float* Wq_w     = (const float*)d_in[5];
  const float* Wq_b     = (const float*)d_in[6];
  const float* Wk_w     = (const float*)d_in[7];
  const float* Wk_b     = (const float*)d_in[8];
  const float* rule_emb = (const float*)d_in[9];
  const float* bias     = (const float*)d_in[10];
  const float* ln_g     = (const float*)d_in[11];
  const float* ln_b     = (const float*)d_in[12];
  float* out = (float*)d_out;

  const int n_nodes = in_sizes[0] / IN_DIM;  // 50000
  const int n_edges = in_sizes[2];           // 800000
  const size_t node64 = (size_t)n_nodes * 64;

  // workspace layout (floats); atomically-updated region contiguous for one zero pass
  float* w = (float*)d_ws;
  size_t off = 0;
  float* ws_q   = w + off; off += node64;                 // q
  float* ws_xk  = w + off; off += node64;                 // x @ Wk1
  float* ws_xw  = w + off; off += (size_t)N_REL * node64; // x @ W_r[r]
  float* ws_qr  = w + off; off += (size_t)n_nodes * 4;    // q . key_rule
  float* ws_es  = w + off; off += (size_t)n_edges * 4;    // edge scores / exp
  float* ws_max = w + off; off += (size_t)n_nodes * 4;    // encoded seg max   } zero
  float* ws_sum = w + off; off += (size_t)n_nodes * 4;    // seg exp sum       } zero
  float* ws_acc = w + off; off += node64;                 // output accum      } zero
  size_t zcount = (size_t)n_nodes * 4 * 2 + node64;
  float* ws_relk = w + off; off += N_REL * OUT_DIM;       // relk (+Wk_bias)
  float* ws_kr   = w + off; off += N_ACTIVE * OUT_DIM;    // key_rule

  zero_kernel<<<(int)((zcount + 255) / 256), 256, 0, stream>>>(ws_max, (int)zcount);
  precompute_kernel<<<1, 512, 0, stream>>>(W_r, Wk_w, Wk_b, rule_emb, rule_ids,
                                           ws_relk, ws_kr);

  dim3 ggrid((n_nodes + 127) / 128, 10);  // 8 waves/block * 16 rows/wave; 10 weight mats
  gemm64_wmma_kernel<<<ggrid, 256, 0, stream>>>(x, Wq_w, Wq_b, Wk_w, W_r,
                                                ws_q, ws_xk, ws_xw, n_nodes);

  qr_kernel<<<(n_nodes + 255) / 256, 256, 0, stream>>>(ws_q, ws_kr, ws_qr, n_nodes);

  int egrid = (n_edges + 255) / 256;
  score_kernel<<<egrid, 256, 0, stream>>>(ei, et, ws_q, ws_xk, ws_relk, ws_qr,
                                          ws_es, (unsigned*)ws_max, n_edges);
  expsum_kernel<<<egrid, 256, 0, stream>>>(ei, (const unsigned*)ws_max, ws_es, ws_sum,
                                           n_edges);
  accum_kernel<<<egrid, 256, 0, stream>>>(ei, et, ws_es, ws_sum, ws_xw, ws_acc,
                                          n_edges, n_nodes);

  finalize_kernel<<<(n_nodes * 32 + 255) / 256, 256, 0, stream>>>(ws_acc, bias, ln_g, ln_b,
                                                                  out, n_nodes);
}